// GCP2_37847251812663
// MI455X (gfx1250) — compile-verified
//
#include <hip/hip_runtime.h>
#include <math.h>

// ---------------------------------------------------------------------------
// GCP layer for MI455X (gfx1250, wave32).
// Edge scatter factors out: sh/gv only need Fsum[n] = sum of frames over edges
// with row==n, plus deg[n].  One atomic edge pass + one per-node pass.
// The 16x153 @ 153x128 node GEMM runs on v_wmma_f32_16x16x4_f32 (full f32).
// W_so is pre-padded to [160][128] in workspace so the WMMA loop has zero
// branches: 1 ds_load_b64 (A) + 16 global_load_b32 (B, L2-hot) + 8 WMMA.
// ---------------------------------------------------------------------------

typedef float v2f __attribute__((ext_vector_type(2)));
typedef float v4f __attribute__((ext_vector_type(4)));
typedef float v8f __attribute__((ext_vector_type(8)));

#define SI 128
#define SO 128
#define VI 16
#define VO 16
#define HD 16
#define MERGED 153          // SI + HD + 9
#define KPAD 160            // 40 k-chunks of 4
#define MSTR 164            // merged LDS row stride (164*m mod 64 banks all distinct)
#define SGSTR 132           // sigmoid LDS row stride (132*m mod 64 distinct for m<16)
#define GEPS 1e-8f

__global__ void gcp_zero(float* acc, int n10) {
  int i = blockIdx.x * blockDim.x + threadIdx.x;
  if (i < n10) acc[i] = 0.0f;
}

// Wp[160][128]: rows 0..152 = W_so, rows 153..159 = 0
__global__ void gcp_padW(const float* __restrict__ W_so, float* __restrict__ Wp) {
  int i = blockIdx.x * blockDim.x + threadIdx.x;
  if (i < KPAD * SO) Wp[i] = (i < MERGED * SO) ? W_so[i] : 0.0f;
}

__global__ void gcp_edge(const int* __restrict__ row,
                         const float* __restrict__ frames,
                         float* __restrict__ acc, int E) {
  int e = blockIdx.x * blockDim.x + threadIdx.x;
  if (e >= E) return;
  int n = row[e];
  const float* f = frames + (size_t)e * 9;
  float* a = acc + (size_t)n * 10;
#pragma unroll
  for (int j = 0; j < 9; ++j) atomicAdd(a + j, f[j]);
  atomicAdd(a + 9, 1.0f);
}

__global__ void __launch_bounds__(32)
gcp_node(const float* __restrict__ s,
         const float* __restrict__ v,
         const float* __restrict__ W_vd,
         const float* __restrict__ W_vdf,
         const float* __restrict__ Wp,       // padded W_so [160][128]
         const float* __restrict__ b_so,
         const float* __restrict__ W_vu,
         const float* __restrict__ W_vosf,
         const float* __restrict__ b_vosf,
         const float* __restrict__ W_vuf,
         const float* __restrict__ nodeAcc,
         float* __restrict__ out_s,
         float* __restrict__ out_v,
         int N) {
  __shared__ float smA[16 * MSTR];     // merged [16][160] padded
  __shared__ float smSig[16 * SGSTR];  // sigmoid(scalar_rep)
  __shared__ float smVr[16 * 48];      // vr_pre per node, [o*3+d]

  const int lane = threadIdx.x;
  const int m    = lane & 15;
  const int hi   = lane >> 4;
  const int base = blockIdx.x * 16;

  // stage s rows into merged cols 0..127 as float4 (coalesced), zero K padding
  for (int r = 0; r < 16; ++r) {
    int n = base + r; if (n >= N) n = N - 1;
    const v4f* sp = (const v4f*)(s + (size_t)n * SI);
    *(v4f*)&smA[r * MSTR + lane * 4] = sp[lane];
  }
  for (int idx = lane; idx < 16 * (MSTR - MERGED); idx += 32) {
    int r = idx / (MSTR - MERGED);
    int c = MERGED + idx % (MSTR - MERGED);
    smA[r * MSTR + c] = 0.0f;
  }

  float Fb[3][3];  // live only in lanes 0..15

  if (hi == 0) {   // per-node prep: lane m owns node base+m
    int n = base + m; if (n >= N) n = N - 1;
    const float* vp = v + (size_t)n * VI * 3;
    float vv[VI][3];
#pragma unroll
    for (int i = 0; i < VI; ++i) {
      vv[i][0] = vp[i*3+0]; vv[i][1] = vp[i*3+1]; vv[i][2] = vp[i*3+2];
    }
    // vh[d][h] = v_pre @ W_vd
    float vh[3][HD];
#pragma unroll
    for (int d = 0; d < 3; ++d)
#pragma unroll
      for (int h = 0; h < HD; ++h) vh[d][h] = 0.0f;
#pragma unroll
    for (int i = 0; i < VI; ++i)
#pragma unroll
      for (int h = 0; h < HD; ++h) {
        float w = W_vd[i * HD + h];
        vh[0][h] += vv[i][0] * w;
        vh[1][h] += vv[i][1] * w;
        vh[2][h] += vv[i][2] * w;
      }
    // vnorm -> merged cols 128..143
#pragma unroll
    for (int h = 0; h < HD; ++h) {
      float t = vh[0][h]*vh[0][h] + vh[1][h]*vh[1][h] + vh[2][h]*vh[2][h];
      smA[m * MSTR + SI + h] = sqrtf(t + GEPS);
    }
    // vdf[d][c] = v_pre @ W_vdf
    float vdf[3][3];
#pragma unroll
    for (int d = 0; d < 3; ++d) { vdf[d][0] = vdf[d][1] = vdf[d][2] = 0.0f; }
#pragma unroll
    for (int i = 0; i < VI; ++i)
#pragma unroll
      for (int c = 0; c < 3; ++c) {
        float w = W_vdf[i*3+c];
        vdf[0][c] += vv[i][0]*w; vdf[1][c] += vv[i][1]*w; vdf[2][c] += vv[i][2]*w;
      }
    // Fbar = inv_deg * sum(frames)
    const float* na = nodeAcc + (size_t)n * 10;
    float inv = 1.0f / fmaxf(na[9], 1.0f);
#pragma unroll
    for (int i2 = 0; i2 < 3; ++i2)
#pragma unroll
      for (int j = 0; j < 3; ++j) Fb[i2][j] = na[i2*3+j] * inv;
    // sh[c*3+i] = (Fbar @ vdf)[i][c] -> merged cols 144..152
#pragma unroll
    for (int i2 = 0; i2 < 3; ++i2)
#pragma unroll
      for (int c = 0; c < 3; ++c)
        smA[m * MSTR + SI + HD + c*3 + i2] =
            Fb[i2][0]*vdf[0][c] + Fb[i2][1]*vdf[1][c] + Fb[i2][2]*vdf[2][c];
    // vr_pre[d][o] = vh @ W_vu, stored [o*3+d]
#pragma unroll
    for (int o = 0; o < VO; ++o) {
      float r0 = 0, r1 = 0, r2 = 0;
#pragma unroll
      for (int h = 0; h < HD; ++h) {
        float w = W_vu[h * VO + o];
        r0 += vh[0][h]*w; r1 += vh[1][h]*w; r2 += vh[2][h]*w;
      }
      smVr[m*48 + o*3 + 0] = r0;
      smVr[m*48 + o*3 + 1] = r1;
      smVr[m*48 + o*3 + 2] = r2;
    }
  }
  __syncthreads();

  // ---- [16 x 160] @ [160 x 128] via V_WMMA_F32_16X16X4_F32 ----
  // A frag (16x4 f32): lane m<16 -> M=m, float2 = A[m][k0+2*hi .. +1]
  // B frag (4x16 f32): lane -> N=m, float2 = B[k0+2*hi][n], B[k0+2*hi+1][n]
  // C/D (16x16 f32): 8 VGPRs, elem r -> M = r + 8*hi, N = m
  v8f acc[8] = {};
  const float* WpRow0 = Wp + (size_t)(2 * hi) * SO + m;  // row k0+2*hi, col base m
  for (int kc = 0; kc < KPAD / 4; ++kc) {
    int k0 = kc * 4;
    v2f a = *(const v2f*)&smA[m * MSTR + k0 + 2*hi];
    const float* wr = WpRow0 + (size_t)k0 * SO;
#pragma unroll
    for (int t = 0; t < 8; ++t) {
      v2f b;
      b.x = wr[t * 16];            // B[k0+2*hi  ][t*16+m]
      b.y = wr[SO + t * 16];       // B[k0+2*hi+1][t*16+m]
      acc[t] = __builtin_amdgcn_wmma_f32_16x16x4_f32(
          false, a, false, b, (short)0, acc[t], false, false);
    }
  }

  // ---- bias, ReLU output, sigmoid into LDS ----
#pragma unroll
  for (int t = 0; t < 8; ++t) {
    int col = t*16 + m;
    float bias = b_so[col];
#pragma unroll
    for (int r = 0; r < 8; ++r) {
      int nr = r + 8*hi;
      int n = base + nr;
      float val = acc[t][r] + bias;
      if (n < N) out_s[(size_t)n * SO + col] = fmaxf(val, 0.0f);
      smSig[nr * SGSTR + col] = 1.0f / (1.0f + expf(-val));
    }
  }
  __syncthreads();

  // ---- gate GEMV + frame-gated vector output (lane m owns node base+m) ----
  if (hi == 0) {
    int n = base + m;
    float g[9];
#pragma unroll
    for (int j = 0; j < 9; ++j) g[j] = b_vosf[j];
    for (int c = 0; c < SO; ++c) {
      float sc = smSig[m * SGSTR + c];
#pragma unroll
      for (int j = 0; j < 9; ++j) g[j] += sc * W_vosf[c*9 + j];
    }
    // gv[i][k] = sum_j gate[j*3+i] * Fbar[j][k]
    float gv[3][3];
#pragma unroll
    for (int i2 = 0; i2 < 3; ++i2)
#pragma unroll
      for (int k = 0; k < 3; ++k)
        gv[i2][k] = g[0+i2]*Fb[0][k] + g[3+i2]*Fb[1][k] + g[6+i2]*Fb[2][k];
    if (n < N) {
      float* ov = out_v + (size_t)n * (VO*3);
#pragma unroll
      for (int o = 0; o < VO; ++o) {
        float w0 = W_vuf[0*VO+o], w1 = W_vuf[1*VO+o], w2 = W_vuf[2*VO+o];
        float x0 = gv[0][0]*w0 + gv[1][0]*w1 + gv[2][0]*w2;
        float x1 = gv[0][1]*w0 + gv[1][1]*w1 + gv[2][1]*w2;
        float x2 = gv[0][2]*w0 + gv[1][2]*w1 + gv[2][2]*w2;
        float gn = sqrtf(x0*x0 + x1*x1 + x2*x2 + GEPS);
        float scale = 1.0f / (1.0f + expf(-gn));
        ov[o*3+0] = smVr[m*48 + o*3 + 0] * scale;
        ov[o*3+1] = smVr[m*48 + o*3 + 1] * scale;
        ov[o*3+2] = smVr[m*48 + o*3 + 2] * scale;
      }
    }
  }
}

extern "C" void kernel_launch(void* const* d_in, const int* in_sizes, int n_in,
                              void* d_out, int out_size, void* d_ws, size_t ws_size,
                              hipStream_t stream) {
  const float* s      = (const float*)d_in[0];
  const float* v      = (const float*)d_in[1];
  const float* frames = (const float*)d_in[2];
  const float* W_vd   = (const float*)d_in[3];
  const float* W_vdf  = (const float*)d_in[4];
  const float* W_so   = (const float*)d_in[5];
  const float* b_so   = (const float*)d_in[6];
  const float* W_vu   = (const float*)d_in[7];
  const float* W_vosf = (const float*)d_in[8];
  const float* b_vosf = (const float*)d_in[9];
  const float* W_vuf  = (const float*)d_in[10];
  const int*   eidx   = (const int*)d_in[11];

  const int N = in_sizes[0] / SI;
  const int E = in_sizes[2] / 9;

  float* nodeAcc = (float*)d_ws;                      // N*10 floats (Fsum[9]+deg)
  float* Wp      = nodeAcc + (size_t)N * 10;          // padded W_so [160][128]
  float* out_s   = (float*)d_out;                     // [N,128]
  float* out_v   = out_s + (size_t)N * SO;            // [N,16,3]

  const int n10 = N * 10;
  gcp_zero<<<(n10 + 255) / 256, 256, 0, stream>>>(nodeAcc, n10);
  gcp_padW<<<(KPAD * SO + 255) / 256, 256, 0, stream>>>(W_so, Wp);
  gcp_edge<<<(E + 255) / 256, 256, 0, stream>>>(eidx, frames, nodeAcc, E);
  gcp_node<<<(N + 15) / 16, 32, 0, stream>>>(s, v, W_vd, W_vdf, Wp, b_so,
                                             W_vu, W_vosf, b_vosf, W_vuf,
                                             nodeAcc, out_s, out_v, N);
}